// MILoss_56040733278514
// MI455X (gfx1250) — compile-verified
//
#include <hip/hip_runtime.h>
#include <math.h>

// ---------------- problem constants (from reference) ----------------
#define TEMP_INV   10.0f          // 1 / TEMPERATURE
#define EPS        1e-8f
#define L_LAYERS   12
#define BS         2048           // B * S
#define D_DIM      1024
#define N_ANCH     2048
#define K_NEG      5
#define NPAIR      (L_LAYERS * N_ANCH)   // 24576 (l,n) pairs
#define NBLK       (NPAIR / 2)           // 12288 blocks, 2 pairs per wave
#define ROW_STRIDE 1028           // floats; 1024+4 pad -> distinct LDS banks per row
#define NROWS      14             // 7 rows per (l,n) pair, 2 pairs per wave
#define PAD_ROW    14             // extra all-zero row read by WMMA pad lanes
#define NROWS_TOT  15

typedef __attribute__((ext_vector_type(2))) float v2f;
typedef __attribute__((ext_vector_type(8))) float v8f;

// Pointer types matching the async-to-LDS builtin's parameters:
// (int4 addrspace(1)* src, int4 addrspace(3)* dst, imm offset, imm cpol)
typedef int i32x4 __attribute__((vector_size(16)));
typedef i32x4 __attribute__((address_space(1)))* as1_i32x4_p;
typedef i32x4 __attribute__((address_space(3)))* as3_i32x4_p;

// gfx1250 async global->LDS path (ASYNCcnt), with plain-copy fallback.
#if defined(__has_builtin)
#if __has_builtin(__builtin_amdgcn_global_load_async_to_lds_b128)
#define USE_ASYNC_LDS 1
#endif
#endif

__device__ __forceinline__ void wait_asynccnt0() {
#if defined(__has_builtin) && __has_builtin(__builtin_amdgcn_s_wait_asynccnt)
    __builtin_amdgcn_s_wait_asynccnt(0);
#else
    asm volatile("s_wait_asynccnt 0x0" ::: "memory");
#endif
}

// Gram-matrix kernel: one wave32 per two (l,n) pairs.
// LDS X holds 14 gathered rows (pair0 -> rows 0..6, pair1 -> 7..13) plus one
// zero pad row. Logical 16x16 WMMA A row m maps: m 0..6 -> pair0, 8..14 ->
// pair1, m==7/m==15 -> pad row (zeros) so the inner loop is branchless.
__global__ __launch_bounds__(32) void mi_gram_kernel(
    const float* __restrict__ act,
    const int*   __restrict__ anchor_idx,
    const int*   __restrict__ pos_idx,
    const int*   __restrict__ neg_idx,
    float*       __restrict__ terms)
{
    __shared__ float X[NROWS_TOT * ROW_STRIDE];  // 61,680 B
    __shared__ float gram[16][17];               // +pad col: no bank conflicts

    const int lane = threadIdx.x;                // block == one wave32
    const int g0   = blockIdx.x * 2;             // first (l,n) pair id
    const int l    = g0 / N_ANCH;                // both pairs share a layer
    const int n0   = g0 % N_ANCH;
    const int n1   = n0 + 1;

    // ---- source row indices for the 14 staged rows (wave-uniform) ----
    int rows[NROWS];
    rows[0] = anchor_idx[n0];
    rows[1] = pos_idx[n0];
#pragma unroll
    for (int k = 0; k < K_NEG; ++k) rows[2 + k] = neg_idx[n0 * K_NEG + k];
    rows[7] = anchor_idx[n1];
    rows[8] = pos_idx[n1];
#pragma unroll
    for (int k = 0; k < K_NEG; ++k) rows[9 + k] = neg_idx[n1 * K_NEG + k];

    const size_t layer_base = (size_t)l * (size_t)BS * (size_t)D_DIM;

    // ---- zero the pad row (ds stores; disjoint from async-written rows) ----
    {
        const float4 z = make_float4(0.f, 0.f, 0.f, 0.f);
        float4* prow = (float4*)&X[PAD_ROW * ROW_STRIDE];
#pragma unroll
        for (int t = lane; t < D_DIM / 4; t += 32) prow[t] = z;
    }

    // ---- stage 14 rows (4 KB each) into LDS ----
#pragma unroll
    for (int r = 0; r < NROWS; ++r) {
        const float4* __restrict__ src =
            (const float4*)(act + layer_base + (size_t)rows[r] * D_DIM);
        float4* dst = (float4*)&X[r * ROW_STRIDE];   // 4112 B stride, 16B aligned
#pragma unroll
        for (int t = lane; t < D_DIM / 4; t += 32) {
#ifdef USE_ASYNC_LDS
            __builtin_amdgcn_global_load_async_to_lds_b128(
                (as1_i32x4_p)(src + t), (as3_i32x4_p)(dst + t),
                /*offset=*/0, /*cpol=*/0);
#else
            dst[t] = src[t];
#endif
        }
    }
#ifdef USE_ASYNC_LDS
    wait_asynccnt0();
#endif
    __syncthreads();

    // ---- accumulate 16x16 fp32 Gram over K = 1024 in steps of 4 ----
    // A 16x4 f32 fragment: lane (m = lane&15, half = lane>>4), vgpr v holds
    // element (row m, k = kb + 2*half + v). B = A^T of the same data, so one
    // float2 per lane feeds both operands. Pad lanes read the zero row
    // (same-address LDS reads broadcast; pad row sits on an unused bank).
    const int m    = lane & 15;
    const int half = lane >> 4;
    const int srow = (m == 7 || m == 15) ? PAD_ROW : ((m < 7) ? m : (m - 1));
    const float* xrow = &X[srow * ROW_STRIDE] + half * 2;

    v8f acc = {0.f, 0.f, 0.f, 0.f, 0.f, 0.f, 0.f, 0.f};

    // software-pipeline one fragment ahead so ds_load latency overlaps WMMA
    v2f frag = *(const v2f*)(xrow);
#pragma unroll 8
    for (int kb = 4; kb < D_DIM; kb += 4) {
        const v2f nxt = *(const v2f*)(xrow + kb);          // ds_load_b64
        acc = __builtin_amdgcn_wmma_f32_16x16x4_f32(
            false, frag, false, frag, (short)0, acc, false, false);
        frag = nxt;
    }
    acc = __builtin_amdgcn_wmma_f32_16x16x4_f32(
        false, frag, false, frag, (short)0, acc, false, false);

    // ---- scatter D into LDS gram: VGPR j, lanes 0-15 -> (M=j, N=lane),
    //      lanes 16-31 -> (M=8+j, N=lane-16) ----
    if (lane < 16) {
#pragma unroll
        for (int j = 0; j < 8; ++j) gram[j][lane] = acc[j];
    } else {
#pragma unroll
        for (int j = 0; j < 8; ++j) gram[8 + j][lane - 16] = acc[j];
    }
    __syncthreads();

    // ---- lanes 0 and 1 finish the two InfoNCE terms ----
    if (lane < 2) {
        const int b = 8 * lane;                 // Gram block base for this pair
        const float na  = fmaxf(sqrtf(gram[b][b]),         EPS);
        const float npn = fmaxf(sqrtf(gram[b + 1][b + 1]), EPS);
        const float pos = gram[b][b + 1] / (na * npn) * TEMP_INV;
        float denom = __expf(pos);
#pragma unroll
        for (int k = 0; k < K_NEG; ++k) {
            const int j = b + 2 + k;
            const float nn = fmaxf(sqrtf(gram[j][j]), EPS);
            const float ns = gram[b][j] / (na * nn) * TEMP_INV;
            denom += __expf(ns);
        }
        terms[g0 + lane] = pos - __logf(denom); // log(numer/denom)
    }
}

// Deterministic single-block tree reduction of the 24576 per-pair terms.
__global__ __launch_bounds__(256) void mi_reduce_kernel(
    const float* __restrict__ terms, float* __restrict__ out)
{
    __shared__ float sm[256];
    float s = 0.f;
    for (int i = threadIdx.x; i < NPAIR; i += 256) s += terms[i];
    sm[threadIdx.x] = s;
    __syncthreads();
#pragma unroll
    for (int w = 128; w > 0; w >>= 1) {
        if ((int)threadIdx.x < w) sm[threadIdx.x] += sm[threadIdx.x + w];
        __syncthreads();
    }
    if (threadIdx.x == 0) out[0] = -sm[0] / (float)NPAIR;
}

extern "C" void kernel_launch(void* const* d_in, const int* in_sizes, int n_in,
                              void* d_out, int out_size, void* d_ws, size_t ws_size,
                              hipStream_t stream)
{
    const float* act    = (const float*)d_in[0];   // (L,B,S,D) fp32
    const int*   anchor = (const int*)d_in[1];     // (N,)
    const int*   pos    = (const int*)d_in[2];     // (N,)
    const int*   neg    = (const int*)d_in[3];     // (N,K)
    float*       out    = (float*)d_out;           // scalar
    float*       terms  = (float*)d_ws;            // NPAIR floats = 96 KB scratch

    mi_gram_kernel<<<NBLK, 32, 0, stream>>>(act, anchor, pos, neg, terms);
    mi_reduce_kernel<<<1, 256, 0, stream>>>(terms, out);
}